// AcidSynth_62165356642838
// MI455X (gfx1250) — compile-verified
//
#include <hip/hip_runtime.h>
#include <hip/hip_bf16.h>
#include <math.h>

// ---------------------------------------------------------------------------
// AcidSynth dry path for MI455X (gfx1250, wave32).
//
// out[i] = osc_gain * env[i] * (1 - shape/2) * tanh(pi*partials/2 * sin(c*(i+1)))
//                               * (1 + shape * cos(c*(i+1)))
// env[i] = 1 - (i % nos)*slope, with a linspace tail on the last (N % nos).
//
// Memory-bound: ~16 MiB of stores, ~16 B of loads (dead inputs never touched).
// Roofline: 16 MiB / 23.3 TB/s ~= 0.7 us.  sin/cos for each 256-sample tile
// come from two V_WMMA_F32_16X16X4_F32 via the angle-addition identity
// (16x2 . 2x16 zero-padded to K=4): 2 transcendentals per 16 samples instead
// of 2 per sample.  tanh is the gfx1250 hardware V_TANH_F32 (TRANS32).
// Tile index is readfirstlane-scalarized so all per-tile control flow and the
// envelope modulus stay in SALU (incremental mod, no per-tile division).
// ---------------------------------------------------------------------------

typedef float v2f __attribute__((ext_vector_type(2)));
typedef float v8f __attribute__((ext_vector_type(8)));

__device__ __forceinline__ float fast_tanh(float x) {
#if __has_builtin(__builtin_amdgcn_tanhf)
    return __builtin_amdgcn_tanhf(x);           // v_tanh_f32, 1 TRANS op
#else
    // Branch-free fallback: tanh(|x|) = 1 - 2/(exp2(2|x|*log2e) + 1).
    float ax = __builtin_fabsf(x);
    float e  = __builtin_amdgcn_exp2f(ax * 2.8853900817779268f); // 2*log2(e)
    float t  = 1.0f - 2.0f * __builtin_amdgcn_rcpf(e + 1.0f);
    return __builtin_copysignf(t, x);
#endif
}

__global__ __launch_bounds__(256) void acid_dry_kernel(
    const float* __restrict__ midi_f0_0to1,        // d_in[1], 1 elem
    const float* __restrict__ note_on_dur_0to1,    // d_in[2], 1 elem
    const float* __restrict__ osc_shape_p,         // d_in[5], 1 elem
    const float* __restrict__ osc_gain_p,          // d_in[6], 1 elem
    float* __restrict__ out, int N)
{
    // ---- uniform scalar derivation (matches Python host-side semantics) ----
    double nd  = (double)note_on_dur_0to1[0];
    double dur = nd * (0.5 - 0.125) + 0.125;
    int    nos = (int)(dur * 48000.0);                    // int() truncation
    double midi = rint((double)midi_f0_0to1[0] * 30.0 + 30.0);  // round-half-even
    double f0   = 440.0 * exp2((midi - 69.0) / 12.0);

    float c        = (float)(6.283185307179586 * f0 / 48000.0);  // rad/sample
    float partials = (float)(48000.0 / (2.0 * f0));
    float kp       = 3.14159265f * partials * 0.5f;              // tanh drive

    double slope_d = 1.0 / (double)(nos - 1);
    float  slope   = (float)slope_d;

    int   r          = N % nos;
    int   tail_start = N - r;
    float tail_step  = 0.0f;
    if (r > 1) {
        double endv = 1.0 - (double)r * slope_d;
        if (endv < 0.0) endv = 0.0;
        tail_step = (float)((endv - 1.0) / (double)(r - 1));     // linspace step
    }

    float shape = osc_shape_p[0];
    float gain  = osc_gain_p[0];
    float g2    = gain * (1.0f - shape * 0.5f);    // folded output scale

    // ---- wave / lane decomposition (tile index scalarized) ----
    const int lane   = (int)(threadIdx.x & 31u);
    const int nlane  = lane & 15;
    const int half   = lane >> 4;                  // 0: lanes 0-15, 1: 16-31
    const bool lo    = (half == 0);
    const int wPerBlk = (int)(blockDim.x >> 5);
    // Wave id is uniform within a wave; force it into an SGPR so the tile
    // loop, bounds checks and envelope modulus all run on the SALU.
    int gw = __builtin_amdgcn_readfirstlane(
                 (int)(blockIdx.x * wPerBlk + (threadIdx.x >> 5)));
    const int nw = (int)(gridDim.x * wPerBlk);

    // ---- constant B matrices (4x16): angle-addition offsets cos(c*n), sin(c*n)
    // Layout: lanes 0-15 carry rows K=0 (Vgpr0) and K=1 (Vgpr1);
    // lanes 16-31 carry K=2,3 which we zero (K padded 2->4).
    float cn = c * (float)nlane;
    float sb = sinf(cn), cb = cosf(cn);
    v2f Bs, Bc;
    Bs.x = lo ?  cb : 0.0f;   // D_sin = sinA*cosB + cosA*sinB
    Bs.y = lo ?  sb : 0.0f;
    Bc.x = lo ? -sb : 0.0f;   // D_cos = cosA*cosB - sinA*sinB
    Bc.y = lo ?  cb : 0.0f;

    const int numTiles = (N + 255) >> 8;           // 256 samples per tile

    // Incremental modulus: one division for the start, one for the stride,
    // then scalar add + conditional subtract per tile (stepMod < nos).
    int baseMod = (gw << 8) % nos;
    int stepMod = (int)(((long long)nw << 8) % (long long)nos);

    const int off0 = 128 * half + nlane;           // C/D layout -> sample offset

    for (int tile = gw; tile < numTiles; tile += nw) {
        const int base = tile << 8;                // scalar tile start sample

        // A (16x4): row M = [sin(c*(base+1+16M)), cos(c*(base+1+16M)), 0, 0]
        float tb = (float)(base + 1 + 16 * nlane);
        float ab = c * tb;
        float sa = sinf(ab), ca = cosf(ab);
        v2f A;
        A.x = lo ? sa : 0.0f;
        A.y = lo ? ca : 0.0f;

        v8f zero = {};
        // D[M][N] = sin / cos of c*(base + 1 + 16M + N) for the whole tile
        v8f S  = __builtin_amdgcn_wmma_f32_16x16x4_f32(
                     false, A, false, Bs, (short)0, zero, false, false);
        v8f Co = __builtin_amdgcn_wmma_f32_16x16x4_f32(
                     false, A, false, Bc, (short)0, zero, false, false);

        const int bm = baseMod;
        if (base + 256 <= N) {
            // Hot path: full tile, scalar-branch entry, branch-free body.
            #pragma unroll
            for (int v = 0; v < 8; ++v) {
                int off = off0 + 16 * v;
                int idx = base + off;
                int mod = bm + off;
                if (mod >= nos) mod -= nos;                      // v_cndmask
                float env  = 1.0f - (float)mod * slope;
                float envT = fmaf((float)(idx - tail_start), tail_step, 1.0f);
                env = (idx >= tail_start) ? envT : env;          // v_cndmask
                float sq  = fast_tanh(kp * S[v]);                // v_tanh_f32
                float osc = sq * fmaf(shape, Co[v], 1.0f);
                out[idx] = g2 * env * osc;
            }
        } else {
            // Cold path: final partial tile only.
            #pragma unroll
            for (int v = 0; v < 8; ++v) {
                int off = off0 + 16 * v;
                int idx = base + off;
                if (idx < N) {
                    int mod = bm + off;
                    if (mod >= nos) mod -= nos;
                    float env  = 1.0f - (float)mod * slope;
                    float envT = fmaf((float)(idx - tail_start), tail_step, 1.0f);
                    env = (idx >= tail_start) ? envT : env;
                    float sq  = fast_tanh(kp * S[v]);
                    float osc = sq * fmaf(shape, Co[v], 1.0f);
                    out[idx] = g2 * env * osc;
                }
            }
        }

        baseMod += stepMod;                        // scalar incremental mod
        if (baseMod >= nos) baseMod -= nos;
    }
}

extern "C" void kernel_launch(void* const* d_in, const int* in_sizes, int n_in,
                              void* d_out, int out_size, void* d_ws, size_t ws_size,
                              hipStream_t stream) {
    (void)in_sizes; (void)n_in; (void)d_ws; (void)ws_size;
    // setup_inputs order: 0:x 1:midi_f0 2:note_on_dur 3:w_mod 4:q_mod
    //                     5:osc_shape 6:osc_gain 7:dist_gain
    const float* midi  = (const float*)d_in[1];
    const float* dur   = (const float*)d_in[2];
    const float* shape = (const float*)d_in[5];
    const float* gain  = (const float*)d_in[6];
    float* out = (float*)d_out;

    int N = out_size;                       // 4,194,304 f32 samples
    const int threads = 256;                // 8 waves / block
    int numTiles = (N + 255) / 256;         // one 256-sample tile per wave-iter
    int blocks = (numTiles + (threads / 32) - 1) / (threads / 32);
    if (blocks > 4096) blocks = 4096;       // grid-stride beyond this

    acid_dry_kernel<<<blocks, threads, 0, stream>>>(midi, dur, shape, gain, out, N);
}